// HistogramLoss_27092653703852
// MI455X (gfx1250) — compile-verified
//
#include <hip/hip_runtime.h>
#include <hip/hip_bf16.h>

typedef __attribute__((ext_vector_type(2))) float v2f;
typedef __attribute__((ext_vector_type(8))) float v8f;

#define C_CH 64
#define NPIX 16384   // 128*128
#define BINS 256

// workspace layout (floats):
//  [0,64)    mm_min   (masked match, per channel)
//  [64,128)  mm_max
//  [128,192) x_min    (masked input, per channel)
//  [192,256) x_max
//  [256]     mask_sum
//  [257]     loss_sum accumulator (zeroed each call by k_minmax)
//  [320, 320+C*BINS)        hist
//  [16704, 16704+C*BINS)    cdf
//  [33088, 33088+257)       T gather table
#define WS_MMMIN 0
#define WS_MMMAX 64
#define WS_XMIN  128
#define WS_XMAX  192
#define WS_MSUM  256
#define WS_LOSS  257
#define WS_HIST  320
#define WS_CDF   16704
#define WS_TTAB  33088

// ---------------- K1: per-channel min/max + mask sum ----------------
__global__ void k_minmax(const float* __restrict__ input, const float* __restrict__ match,
                         const float* __restrict__ mask, float* __restrict__ ws) {
  __shared__ float s0[256], s1[256], s2[256], s3[256], s4[256];
  const int c = blockIdx.x, t = threadIdx.x;
  const float* mrow = match + c * NPIX;
  const float* irow = input + c * NPIX;
  float mmmin = 3.4e38f, mmmax = -3.4e38f, xmin = 3.4e38f, xmax = -3.4e38f, msum = 0.f;
  for (int n = t; n < NPIX; n += 256) {
    float mk = mask[n];
    float mv = mrow[n] * mk;
    float xv = irow[n] * mk;
    mmmin = fminf(mmmin, mv); mmmax = fmaxf(mmmax, mv);
    xmin  = fminf(xmin,  xv); xmax  = fmaxf(xmax,  xv);
    msum += mk;
  }
  s0[t] = mmmin; s1[t] = mmmax; s2[t] = xmin; s3[t] = xmax; s4[t] = msum;
  __syncthreads();
  for (int ofs = 128; ofs > 0; ofs >>= 1) {
    if (t < ofs) {
      s0[t] = fminf(s0[t], s0[t + ofs]);
      s1[t] = fmaxf(s1[t], s1[t + ofs]);
      s2[t] = fminf(s2[t], s2[t + ofs]);
      s3[t] = fmaxf(s3[t], s3[t + ofs]);
      s4[t] += s4[t + ofs];
    }
    __syncthreads();
  }
  if (t == 0) {
    ws[WS_MMMIN + c] = s0[0];
    ws[WS_MMMAX + c] = s1[0];
    ws[WS_XMIN  + c] = s2[0];
    ws[WS_XMAX  + c] = s3[0];
    if (c == 0) { ws[WS_MSUM] = s4[0]; ws[WS_LOSS] = 0.f; }
  }
}

// ---------------- K2: per-channel histogram (ds_add_f32 LDS atomics) ----------------
__global__ void k_hist(const float* __restrict__ match, const float* __restrict__ mask,
                       float* __restrict__ ws) {
  __shared__ float h[BINS];
  const int c = blockIdx.x, t = threadIdx.x;  // blockDim == BINS == 256
  h[t] = 0.f;
  __syncthreads();
  const float mn = ws[WS_MMMIN + c], mx = ws[WS_MMMAX + c];
  const float w = (mx - mn) / (float)BINS;
  const float sw = (w > 0.f) ? w : 1.0f;   // torch.histc safe width
  const float inv = 1.0f / sw;
  const float* mrow = match + c * NPIX;
  for (int n = t; n < NPIX; n += 256) {
    float v = mrow[n] * mask[n];
    int b = (int)floorf((v - mn) * inv);
    b = b < 0 ? 0 : (b > BINS - 1 ? BINS - 1 : b);  // rightmost bin inclusive
    atomicAdd(&h[b], 1.0f);
  }
  __syncthreads();
  // sh = hist * N / sum(hist) == hist (sum is always N), write directly
  ws[WS_HIST + c * BINS + t] = h[t];
}

// ---------------- K3: cdf = hist x UpperTriOnes via V_WMMA_F32_16X16X4_F32 ----------------
// one wave (32 lanes) per 16x16 output tile; grid = (BINS/16, C/16)
__global__ __launch_bounds__(32) void k_cumsum_wmma(float* __restrict__ ws) {
  const float* sh = ws + WS_HIST;
  float* cdf = ws + WS_CDF;
  const int lane = threadIdx.x;       // 0..31, EXEC all-1s (no divergence before wmma)
  const int half = lane >> 4;         // 0: lanes 0-15, 1: lanes 16-31
  const int l    = lane & 15;
  const int j0 = blockIdx.x * 16;     // output bin-column tile
  const int m0 = blockIdx.y * 16;     // channel-row tile
  v8f acc = {};
  for (int k = 0; k < BINS; k += 4) {
    // A (16x4 f32, ISA layout): lanes<16 hold K=k+0,k+1; lanes>=16 hold K=k+2,k+3
    const int ka = k + 2 * half;
    v2f a, b;
    a.x = sh[(m0 + l) * BINS + ka + 0];
    a.y = sh[(m0 + l) * BINS + ka + 1];
    // B (4x16 f32): U[r][col] = (r <= col) ? 1 : 0  (inclusive prefix sum)
    b.x = (ka + 0 <= j0 + l) ? 1.0f : 0.0f;
    b.y = (ka + 1 <= j0 + l) ? 1.0f : 0.0f;
    acc = __builtin_amdgcn_wmma_f32_16x16x4_f32(false, a, false, b,
                                                (short)0, acc, false, false);
  }
  // C/D layout: VGPR v -> row m0+v (lanes 0-15) / m0+v+8 (lanes 16-31), col j0+lane%16
#pragma unroll
  for (int v = 0; v < 8; ++v) {
    cdf[(m0 + v + 8 * half) * BINS + (j0 + l)] = acc[v];
  }
}

// ---------------- K4: build T[0..256] (row-0 corrected values, faithful flat gathers) ----
__global__ void k_ttab(float* __restrict__ ws) {
  const float* cdf = ws + WS_CDF;
  __shared__ float c0[BINS + 1];      // cdf flat [0..256]; [256] == cdf[1,0]
  const int t = threadIdx.x;          // blockDim = 288
  if (t < BINS + 1) c0[t] = cdf[t];
  __syncthreads();
  if (t > BINS) return;               // j = 0..256
  const float rank = (float)(t + 1);
  // idx = count of cdf[0,b] < rank over b in [0,256)  (lower bound)
  int lo = 0, hi = BINS;
  while (lo < hi) { int mid = (lo + hi) >> 1; if (c0[mid] < rank) lo = mid + 1; else hi = mid; }
  const int idx = lo;                                 // 0..256
  const float cdfsel  = c0[idx];                      // flat gather (idx==256 -> cdf[1,0])
  const float cdfprev = (idx == 0 || idx == BINS) ? 0.f : c0[idx - 1]; // cdf_prev flat
  float ratio = (rank - cdfprev) / (1e-8f + cdfsel);
  ratio = ratio < 0.f ? 0.f : (ratio > 1.f ? 1.f : ratio);
  const float xmin0 = ws[WS_XMIN], xmax0 = ws[WS_XMAX];
  const float step0 = (xmax0 - xmin0) / (float)BINS;
  // .at[:,-1].set(channel_max) applies to n==NPIX-1; t<=256 << NPIX-1 -> never hit
  ws[WS_TTAB + t] = xmin0 + (ratio + (float)idx) * step0;
}

// ---------------- K5: loss pass (binary search in LDS + gather T) ----------------
__global__ void k_loss(const float* __restrict__ input, const float* __restrict__ mask,
                       float* __restrict__ ws) {
  __shared__ float cdfc[BINS];
  __shared__ float T[BINS + 1];
  __shared__ float red[256];
  const int c = blockIdx.x, t = threadIdx.x;
  cdfc[t] = ws[WS_CDF + c * BINS + t];
  T[t] = ws[WS_TTAB + t];
  if (t == 0) T[BINS] = ws[WS_TTAB + BINS];
  __syncthreads();
  const float* irow = input + c * NPIX;
  float acc = 0.f;
  for (int n = t; n < NPIX; n += 256) {
    if (n + 4096 < NPIX) __builtin_prefetch(irow + n + 4096, 0, 0);  // global_prefetch_b8
    const float x = irow[n] * mask[n];
    const float rank = (float)(n + 1);
    int lo = 0, hi = BINS;
    while (lo < hi) { int mid = (lo + hi) >> 1; if (cdfc[mid] < rank) lo = mid + 1; else hi = mid; }
    const float d = T[lo] - x;
    acc += d * d;
  }
  red[t] = acc;
  __syncthreads();
  for (int ofs = 128; ofs > 0; ofs >>= 1) {
    if (t < ofs) red[t] += red[t + ofs];
    __syncthreads();
  }
  if (t == 0) atomicAdd(&ws[WS_LOSS], red[0]);   // global_atomic_add_f32
}

// ---------------- K6: finalize scalar ----------------
__global__ void k_final(const float* __restrict__ ws, float* __restrict__ out) {
  const float total = (float)C_CH * (float)NPIX;
  float loss = ws[WS_LOSS] / total;                 // mean((corr-x)^2)
  loss = loss * ws[WS_MSUM] * (float)C_CH / total;  // * mask.sum * C / input.size * WEIGHT(1.0)
  out[0] = loss;
}

extern "C" void kernel_launch(void* const* d_in, const int* in_sizes, int n_in,
                              void* d_out, int out_size, void* d_ws, size_t ws_size,
                              hipStream_t stream) {
  const float* input = (const float*)d_in[0];   // (1,64,128,128) f32
  const float* match = (const float*)d_in[1];   // (1,64,128,128) f32
  const float* mask  = (const float*)d_in[2];   // (1,1,128,128)  f32
  // d_in[3] = bins (int scalar) -- reference fixes bins=256; specialized at compile time
  float* ws  = (float*)d_ws;
  float* out = (float*)d_out;

  k_minmax<<<C_CH, 256, 0, stream>>>(input, match, mask, ws);
  k_hist  <<<C_CH, BINS, 0, stream>>>(match, mask, ws);
  k_cumsum_wmma<<<dim3(BINS / 16, C_CH / 16), 32, 0, stream>>>(ws);
  k_ttab  <<<1, 288, 0, stream>>>(ws);
  k_loss  <<<C_CH, 256, 0, stream>>>(input, mask, ws);
  k_final <<<1, 1, 0, stream>>>(ws, out);
}